// ProductQuantization_25477746000028
// MI455X (gfx1250) — compile-verified
//
#include <hip/hip_runtime.h>

// ---------------------------------------------------------------------------
// Product Quantization scoring on MI455X (gfx1250, wave32, WMMA).
//   Step 1: XR = X (32768x768) * rotate^T (768x768)        -> d_ws
//   Step 2: scores[b,m,k] = dot(XR[b, m*24 : m*24+24], codebook[m,k,:])
// fp32 WMMA (V_WMMA_F32_16X16X4_F32) + async global->LDS DMA (ASYNCcnt)
// with double-buffered LDS in the rotation GEMM.
// ---------------------------------------------------------------------------

typedef float v2f __attribute__((ext_vector_type(2)));
typedef float v8f __attribute__((ext_vector_type(8)));

static constexpr int Bn   = 32768;  // documents
static constexpr int EMB  = 768;    // embedding size
static constexpr int Msub = 32;     // subvectors
static constexpr int Kc   = 256;    // centroids per subvector
static constexpr int Dd   = 24;     // dims per subvector

__device__ __forceinline__ v8f wmma_f32(v2f a, v2f b, v8f c) {
  // 8 args: (neg_a, A, neg_b, B, c_mod, C, reuse_a, reuse_b)
  return __builtin_amdgcn_wmma_f32_16x16x4_f32(false, a, false, b,
                                               (short)0, c, false, false);
}

// Async DMA: 16 bytes global -> LDS per lane, tracked by ASYNCcnt.
// GV mode: 64-bit global address in VGPR pair, LDS byte offset in VGPR.
__device__ __forceinline__ void async_copy16(const float* __restrict__ g,
                                             float* l) {
  const unsigned lds = (unsigned)(unsigned long long)l;  // LDS offset = low 32b
  const unsigned long long ga = (unsigned long long)g;
  asm volatile("global_load_async_to_lds_b128 %0, %1, off"
               :
               : "v"(lds), "v"(ga)
               : "memory");
}

__device__ __forceinline__ void wait_async(void) {
  asm volatile("s_wait_asynccnt 0x0" ::: "memory");
}
__device__ __forceinline__ void wait_async8(void) {
  asm volatile("s_wait_asynccnt 0x8" ::: "memory");
}

// ---------------------------------------------------------------------------
// Kernel 1: XR = X * rotate^T.
// Block = 256 threads (8 waves), tile = 128 rows x 128 cols, K-tile = 32.
// Double-buffered LDS: async DMA of tile t+1 overlaps WMMA on tile t.
// ---------------------------------------------------------------------------
__global__ __launch_bounds__(256) void pq_rotate_gemm(
    const float* __restrict__ x, const float* __restrict__ rotate,
    float* __restrict__ xr) {
  constexpr int BK = 32;
  constexpr int LD = 36;   // padded row stride (144 B: 16B-aligned, no conflicts)
  constexpr int NT = EMB / BK;  // 24 K-tiles
  __shared__ __align__(16) float As[2][128 * LD];
  __shared__ __align__(16) float Bs[2][128 * LD];

  const int tid  = threadIdx.x;
  const int lane = tid & 31;
  const int wave = tid >> 5;
  const int h    = lane >> 4;   // half-wave select (K pair / M half)
  const int l15  = lane & 15;
  const int rowBase = blockIdx.y * 128;
  const int colBase = blockIdx.x * 128;

  // Per-thread slice of the cooperative tile DMA: 8 x b128 per tile.
  const int r0  = tid >> 1;            // even share: rows tid/2 (+64)
  // Mapping: idx = tid + i*256 -> row idx>>3, col4 idx&7 (same as before).
#define ISSUE_TILE(k0, buf)                                                   \
  {                                                                           \
    _Pragma("unroll") for (int i = 0; i < 4; ++i) {                           \
      const int idx = tid + i * 256;                                          \
      const int r   = idx >> 3;                                               \
      const int c4  = idx & 7;                                                \
      async_copy16(x + (size_t)(rowBase + r) * EMB + (k0) + c4 * 4,           \
                   &As[buf][r * LD + c4 * 4]);                                \
      async_copy16(rotate + (size_t)(colBase + r) * EMB + (k0) + c4 * 4,      \
                   &Bs[buf][r * LD + c4 * 4]);                                \
    }                                                                         \
  }

  v8f acc[8] = {};

  ISSUE_TILE(0, 0);
  for (int t = 0; t < NT; ++t) {
    const int cur = t & 1;
    if (t + 1 < NT) {
      // Buffer cur^1 was last consumed in iteration t-1 (barrier below
      // guarantees all waves are done with it) -> safe to refill now.
      ISSUE_TILE((t + 1) * BK, cur ^ 1);
      wait_async8();  // 8 newest outstanding belong to t+1; tile t is done
    } else {
      wait_async();
    }
    __syncthreads();  // all waves' DMA for tile t visible in LDS

#pragma unroll
    for (int kc = 0; kc < BK; kc += 4) {
      // A frag 16x4: lane l15 -> row, half-wave -> K pair {0,1}/{2,3}
      const v2f a =
          *(const v2f*)(&As[cur][(wave * 16 + l15) * LD + kc + h * 2]);
#pragma unroll
      for (int nb = 0; nb < 8; ++nb) {
        // B frag 4x16: B[k][n] = rotate[colBase+n][k]
        const v2f b =
            *(const v2f*)(&Bs[cur][(nb * 16 + l15) * LD + kc + h * 2]);
        acc[nb] = wmma_f32(a, b, acc[nb]);
      }
    }
    __syncthreads();  // done consuming buf cur -> refillable next iteration
  }
#undef ISSUE_TILE
  (void)r0;

  // C layout: VGPR j, lanes 0-15 -> M=j, lanes 16-31 -> M=j+8; N = lane&15.
#pragma unroll
  for (int nb = 0; nb < 8; ++nb) {
#pragma unroll
    for (int j = 0; j < 8; ++j) {
      const int r = rowBase + wave * 16 + h * 8 + j;
      const int c = colBase + nb * 16 + l15;
      xr[(size_t)r * EMB + c] = acc[nb][j];
    }
  }
}

// ---------------------------------------------------------------------------
// Kernel 2: scores[b,m,k] = sum_d XR[b, m*24+d] * codebook[m,k,d].
// Block = 256 threads (8 waves), one m per blockIdx.y,
// tile = 64 rows x 256 centroids; whole D=24 resident in LDS via async DMA.
// ---------------------------------------------------------------------------
__global__ __launch_bounds__(256) void pq_score_gemm(
    const float* __restrict__ xr, const float* __restrict__ codebook,
    float* __restrict__ out) {
  constexpr int LD = 28;  // padded row stride (112 B: 16B-aligned)
  __shared__ __align__(16) float Xs[64 * LD];
  __shared__ __align__(16) float Cs[256 * LD];

  const int tid  = threadIdx.x;
  const int lane = tid & 31;
  const int wave = tid >> 5;
  const int h    = lane >> 4;
  const int l15  = lane & 15;
  const int m       = blockIdx.y;
  const int rowBase = blockIdx.x * 64;
  const int wrow = (wave & 3) * 16;   // 4 row-groups of 16
  const int wcol = (wave >> 2) * 128; // 2 col-groups of 128

  // XR slice: 64 rows x 24 floats = 384 x b128 (EXEC-masked async DMA).
#pragma unroll
  for (int i = 0; i < 2; ++i) {
    const int idx = tid + i * 256;
    if (idx < 384) {
      const int r = idx / 6, c4 = idx % 6;
      async_copy16(xr + (size_t)(rowBase + r) * EMB + m * Dd + c4 * 4,
                   Xs + r * LD + c4 * 4);
    }
  }
  // codebook[m]: 256 rows x 24 floats = 1536 x b128.
#pragma unroll
  for (int i = 0; i < 6; ++i) {
    const int idx = tid + i * 256;
    const int r = idx / 6, c4 = idx % 6;
    async_copy16(codebook + ((size_t)m * Kc + r) * Dd + c4 * 4,
                 Cs + r * LD + c4 * 4);
  }
  wait_async();
  __syncthreads();

  v8f acc[8] = {};
#pragma unroll
  for (int kc = 0; kc < Dd; kc += 4) {
    const v2f a = *(const v2f*)(Xs + (wrow + l15) * LD + kc + h * 2);
#pragma unroll
    for (int nb = 0; nb < 8; ++nb) {
      const v2f b =
          *(const v2f*)(Cs + (wcol + nb * 16 + l15) * LD + kc + h * 2);
      acc[nb] = wmma_f32(a, b, acc[nb]);
    }
  }

  // scores layout: [B, M, K] -> offset b*M*K + m*K + k
#pragma unroll
  for (int nb = 0; nb < 8; ++nb) {
#pragma unroll
    for (int j = 0; j < 8; ++j) {
      const int b = rowBase + wrow + h * 8 + j;
      const int c = wcol + nb * 16 + l15;
      out[((size_t)b * Msub + m) * Kc + c] = acc[nb][j];
    }
  }
}

extern "C" void kernel_launch(void* const* d_in, const int* in_sizes, int n_in,
                              void* d_out, int out_size, void* d_ws,
                              size_t ws_size, hipStream_t stream) {
  (void)in_sizes; (void)n_in; (void)out_size; (void)ws_size;
  const float* x        = (const float*)d_in[0];  // [32768, 768]
  const float* codebook = (const float*)d_in[1];  // [32, 256, 24]
  const float* rotate   = (const float*)d_in[2];  // [768, 768]
  float* out = (float*)d_out;                     // [32768, 32, 256]
  float* xr  = (float*)d_ws;                      // [32768, 768] scratch

  dim3 g1(EMB / 128, Bn / 128);  // (6, 256)
  pq_rotate_gemm<<<g1, 256, 0, stream>>>(x, rotate, xr);

  dim3 g2(Bn / 64, Msub);        // (512, 32)
  pq_score_gemm<<<g2, 256, 0, stream>>>(xr, codebook, out);
}